// VQVAEQuantize_25606595019097
// MI455X (gfx1250) — compile-verified
//
#include <hip/hip_runtime.h>
#include <stdint.h>

// Problem constants (from reference)
#define NROWS   32768      // 4*1024*8 vectors
#define KENT    4096       // codebook entries
#define DIM     64         // entry dim
#define ROWS_WG 128        // rows of z per workgroup (8 waves x 16 rows)
#define KCHUNK  64         // codebook entries per LDS buffer
#define LSTR    68         // DIM + 4 pad: rows are 272B -> b128 stays 16B aligned,
                           // and b64 fragment reads hit 64 distinct LDS banks
#define BUF_FLOATS (KCHUNK * LSTR)   // 4352 floats = 17408 B per buffer

typedef __attribute__((ext_vector_type(2))) float v2f;
typedef __attribute__((ext_vector_type(4))) float v4f;
typedef __attribute__((ext_vector_type(8))) float v8f;

// ---------------------------------------------------------------------------
// Kernel 0: codebook squared norms (4096 x reduce-64)
// ---------------------------------------------------------------------------
__global__ __launch_bounds__(256) void vq_cbnorm_kernel(const float* __restrict__ w,
                                                        float* __restrict__ cbn) {
  int e = blockIdx.x * blockDim.x + threadIdx.x;
  if (e >= KENT) return;
  const v4f* p = (const v4f*)(w + (size_t)e * DIM);
  float s = 0.f;
#pragma unroll
  for (int i = 0; i < DIM / 4; ++i) {
    v4f v = p[i];
    s += v.x * v.x + v.y * v.y + v.z * v.z + v.w * v.w;
  }
  cbn[e] = s;
}

// ---------------------------------------------------------------------------
// Kernel 1: fused distance-GEMM + argmax.
//  - A (16x64 z stripe per wave) held entirely in registers (32 VGPRs).
//  - Codebook streamed through a double-buffered LDS tile using CDNA5
//    GLOBAL_LOAD_ASYNC_TO_LDS_B128 (ASYNCcnt), prefetch overlapped with WMMA.
//  - 4 independent V_WMMA_F32_16X16X4_F32 accumulator chains per chunk.
// ---------------------------------------------------------------------------
__global__ __launch_bounds__(256) void vq_argmax_kernel(const float* __restrict__ z,
                                                        const float* __restrict__ w,
                                                        const float* __restrict__ cbn,
                                                        int* __restrict__ idxOut) {
  __shared__ float smem[2 * BUF_FLOATS];   // 34816 B; doubles as z staging area

  const int tid  = threadIdx.x;
  const int wave = tid >> 5;       // 0..7
  const int lane = tid & 31;
  const int half = lane >> 4;      // 0: K=0,1 / rows M<8 ; 1: K=2,3 / rows M>=8
  const int mn   = lane & 15;      // A: row-in-tile, B/C: col-in-tile
  const int rowBase = blockIdx.x * ROWS_WG;

  // ---- Stage z tile (128 x 64) into smem once, coalesced float4 ----
  {
    const v4f* zg = (const v4f*)(z + (size_t)rowBase * DIM);
#pragma unroll
    for (int i = 0; i < 8; ++i) {
      int linear = tid + i * 256;     // float4 index
      int r = linear >> 4, c4 = linear & 15;
      *(v4f*)&smem[r * LSTR + c4 * 4] = zg[linear];
    }
  }
  __syncthreads();

  // ---- Pull this wave's A stripe into registers (fragment layout:
  //      lane<16 holds K=4kk+0,1 ; lane>=16 holds K=4kk+2,3) ----
  v2f a[16];
  {
    const float* za = &smem[(wave * 16 + mn) * LSTR + 2 * half];
#pragma unroll
    for (int kk = 0; kk < 16; ++kk) a[kk] = *(const v2f*)(za + kk * 4);
  }
  __syncthreads();   // barrier fence drains DScnt; smem reusable as buffers

  // ---- Per-thread async-copy slots: 4 x b128 per 64x64 chunk ----
  const unsigned ldsBase = (unsigned)(uintptr_t)smem;  // low 32 bits = LDS offset
  unsigned ldsOff[4];
  int      lin[4];
#pragma unroll
  for (int i = 0; i < 4; ++i) {
    int linear = tid + i * 256;
    lin[i] = linear;
    int r = linear >> 4, c4 = linear & 15;
    ldsOff[i] = ldsBase + (unsigned)((r * LSTR + c4 * 4) * 4);
  }

  // ---- Prefetch chunk 0 into buffer 0 (async DMA to LDS) ----
#pragma unroll
  for (int i = 0; i < 4; ++i) {
    uint64_t g = (uint64_t)(uintptr_t)w + (uint64_t)lin[i] * 16u;
    unsigned l = ldsOff[i];
    asm volatile("global_load_async_to_lds_b128 %0, %1, off"
                 :: "v"(l), "v"(g) : "memory");
  }
  asm volatile("s_wait_asynccnt 0x0" ::: "memory");
  __syncthreads();

  float best[8];
  int   bidx[8];
#pragma unroll
  for (int v = 0; v < 8; ++v) { best[v] = -3.0e38f; bidx[v] = 0; }

  for (int c = 0; c < KENT / KCHUNK; ++c) {
    const float* cur = &smem[(c & 1) * BUF_FLOATS];

    // Issue async prefetch of next chunk into the other buffer. Safe: that
    // buffer's last readers finished at the barrier ending iteration c-1.
    if (c + 1 < KENT / KCHUNK) {
      unsigned bufSel = (unsigned)(((c + 1) & 1) * (BUF_FLOATS * 4));
      uint64_t gbase  = (uint64_t)(uintptr_t)w +
                        (uint64_t)(c + 1) * (KCHUNK * DIM * 4);
#pragma unroll
      for (int i = 0; i < 4; ++i) {
        uint64_t g = gbase + (uint64_t)lin[i] * 16u;
        unsigned l = ldsOff[i] + bufSel;
        asm volatile("global_load_async_to_lds_b128 %0, %1, off"
                     :: "v"(l), "v"(g) : "memory");
      }
    }

    // ---- 64 WMMA: 4 independent chains, A from registers, B from LDS ----
    v8f acc0 = {}, acc1 = {}, acc2 = {}, acc3 = {};
    const float* b0p = &cur[( 0 + mn) * LSTR + 2 * half];
    const float* b1p = &cur[(16 + mn) * LSTR + 2 * half];
    const float* b2p = &cur[(32 + mn) * LSTR + 2 * half];
    const float* b3p = &cur[(48 + mn) * LSTR + 2 * half];
#pragma unroll
    for (int kk = 0; kk < 16; ++kk) {
      v2f b0 = *(const v2f*)(b0p + kk * 4);
      v2f b1 = *(const v2f*)(b1p + kk * 4);
      v2f b2 = *(const v2f*)(b2p + kk * 4);
      v2f b3 = *(const v2f*)(b3p + kk * 4);
      acc0 = __builtin_amdgcn_wmma_f32_16x16x4_f32(false, a[kk], false, b0, (short)0, acc0, false, false);
      acc1 = __builtin_amdgcn_wmma_f32_16x16x4_f32(false, a[kk], false, b1, (short)0, acc1, false, false);
      acc2 = __builtin_amdgcn_wmma_f32_16x16x4_f32(false, a[kk], false, b2, (short)0, acc2, false, false);
      acc3 = __builtin_amdgcn_wmma_f32_16x16x4_f32(false, a[kk], false, b3, (short)0, acc3, false, false);
    }

    // ---- Running argmax over this chunk's 64 entries ----
#pragma unroll
    for (int ct = 0; ct < 4; ++ct) {
      int e = c * KCHUNK + ct * 16 + mn;
      float cn = cbn[e];
      v8f acc = (ct == 0) ? acc0 : (ct == 1) ? acc1 : (ct == 2) ? acc2 : acc3;
#pragma unroll
      for (int v = 0; v < 8; ++v) {
        float score = 2.0f * acc[v] - cn;
        if (score > best[v]) { best[v] = score; bidx[v] = e; }
      }
    }

    // Drain own async copies, then workgroup-sync before swapping buffers.
    asm volatile("s_wait_asynccnt 0x0" ::: "memory");
    __syncthreads();
  }

  // ---- Reduce argmax over N (16 lanes within each half) ----
#pragma unroll
  for (int v = 0; v < 8; ++v) {
#pragma unroll
    for (int off = 8; off >= 1; off >>= 1) {
      float ov = __shfl_xor(best[v], off, 32);
      int   oi = __shfl_xor(bidx[v], off, 32);
      if (ov > best[v] || (ov == best[v] && oi < bidx[v])) { best[v] = ov; bidx[v] = oi; }
    }
  }
  if (mn == 0) {
    int row = rowBase + wave * 16 + half * 8;   // slot v holds row M = v + 8*half
#pragma unroll
    for (int v = 0; v < 8; ++v) idxOut[row + v] = bidx[v];
  }
}

// ---------------------------------------------------------------------------
// Kernel 2: one wave per row — gather z_q, commitment MSE, histogram,
// embed_sum scatter (f32 global atomics).
// ---------------------------------------------------------------------------
__global__ __launch_bounds__(256) void vq_outputs_kernel(const float* __restrict__ z,
                                                         const float* __restrict__ w,
                                                         const int* __restrict__ idxIn,
                                                         float* __restrict__ outZq,
                                                         float* __restrict__ outIdx,
                                                         float* __restrict__ outCommit,
                                                         float* __restrict__ outEnc,
                                                         float* __restrict__ outEmbed) {
  int wave = threadIdx.x >> 5;
  int lane = threadIdx.x & 31;
  int row  = blockIdx.x * 8 + wave;
  int idx  = idxIn[row];

  v2f zv = *(const v2f*)(z + (size_t)row * DIM + lane * 2);
  v2f wv = *(const v2f*)(w + (size_t)idx * DIM + lane * 2);
  *(v2f*)(outZq + (size_t)row * DIM + lane * 2) = wv;   // straight-through == z_q

  float dx = zv.x - wv.x, dy = zv.y - wv.y;
  float s = dx * dx + dy * dy;
#pragma unroll
  for (int off = 16; off >= 1; off >>= 1) s += __shfl_xor(s, off, 32);

  if (lane == 0) {
    outIdx[row] = (float)idx;
    atomicAdd(&outCommit[row >> 3], s * (1.0f / 512.0f));  // 8 rows per (b,s); /512 dims
    atomicAdd(&outEnc[idx], 1.0f);
  }
  atomicAdd(&outEmbed[(size_t)idx * DIM + lane * 2],     zv.x);
  atomicAdd(&outEmbed[(size_t)idx * DIM + lane * 2 + 1], zv.y);
}

// ---------------------------------------------------------------------------
extern "C" void kernel_launch(void* const* d_in, const int* in_sizes, int n_in,
                              void* d_out, int out_size, void* d_ws, size_t ws_size,
                              hipStream_t stream) {
  const float* z = (const float*)d_in[0];   // (4,1024,8,64) fp32
  const float* w = (const float*)d_in[1];   // (4096,64) fp32

  float* out       = (float*)d_out;
  float* outZq     = out;                               // 2,097,152
  float* outIdx    = outZq + (size_t)NROWS * DIM;       //    32,768
  float* outCommit = outIdx + NROWS;                    //     4,096
  float* outEnc    = outCommit + 4096;                  //     4,096
  float* outEmbed  = outEnc + KENT;                     //   262,144

  float* cbn   = (float*)d_ws;
  int*   idxWs = (int*)((char*)d_ws + KENT * sizeof(float));

  // Accumulated outputs must start at zero every call (harness poisons once).
  hipMemsetAsync(outCommit, 0, (size_t)(4096 + KENT + KENT * DIM) * sizeof(float), stream);

  vq_cbnorm_kernel<<<KENT / 256, 256, 0, stream>>>(w, cbn);
  vq_argmax_kernel<<<NROWS / ROWS_WG, 256, 0, stream>>>(z, w, cbn, idxWs);
  vq_outputs_kernel<<<NROWS / 8, 256, 0, stream>>>(z, w, idxWs, outZq, outIdx,
                                                   outCommit, outEnc, outEmbed);
}